// R2C2_78108275245416
// MI455X (gfx1250) — compile-verified
//
#include <hip/hip_runtime.h>
#include <math.h>

#define T_STEPS 1440
#define CHUNK   (T_STEPS / 32)      // 45 steps per lane
#define ROWS_PER_BLOCK 2            // 2 waves per block, 1 row per wave
#define DT 300.0f

#if defined(__gfx1250__) && __has_builtin(__builtin_amdgcn_global_load_async_to_lds_b128)
#define USE_ASYNC_LDS 1
#else
#define USE_ASYNC_LDS 0
#endif

// Pointee type the async-LDS builtins expect: 16-byte int vector.
typedef int avec16 __attribute__((vector_size(16)));
typedef __attribute__((address_space(1))) avec16* gvec_ptr;
typedef __attribute__((address_space(3))) avec16* lvec_ptr;

__device__ __forceinline__ float sigm(float x) { return 1.0f / (1.0f + expf(-x)); }
__device__ __forceinline__ float bounded(float raw, float lo, float hi) {
    return lo + (hi - lo) * sigm(raw);
}

__global__ __launch_bounds__(ROWS_PER_BLOCK * 32)
void rc2_affine_scan_kernel(const float* __restrict__ in,      // [B, T, 4]
                            const float* __restrict__ rRi,
                            const float* __restrict__ rRe,
                            const float* __restrict__ rCi,
                            const float* __restrict__ rCe,
                            const float* __restrict__ rAi,
                            const float* __restrict__ rAe,
                            const float* __restrict__ rHg,
                            float* __restrict__ out,            // [B, T]
                            int B)
{
    __shared__ float4 s_raw[ROWS_PER_BLOCK][T_STEPS];  // raw (t_in,t_out,hvac,solar)
    __shared__ float  s_o[ROWS_PER_BLOCK][T_STEPS];    // staged outputs

    const int wave = threadIdx.x >> 5;
    const int lane = threadIdx.x & 31;
    const int row  = blockIdx.x * ROWS_PER_BLOCK + wave;
    if (row >= B) return;

    // ---- bounded parameters (uniform scalars -> SALU float path) ----
    const float Ri = bounded(rRi[0], 1e-4f, 0.2f);
    const float Re = bounded(rRe[0], 1e-4f, 0.2f);
    const float Ci = bounded(rCi[0], 1e5f, 1e8f);
    const float Ce = bounded(rCe[0], 1e5f, 1e8f);
    const float Ai = bounded(rAi[0], 0.0f, 0.2f);
    const float Ae = bounded(rAe[0], 0.0f, 0.2f);
    const float Hg = bounded(rHg[0], 1.0f, 20000.0f);
    const float cooling = -1.0f;                      // MODE_COOL

    const float invRi = 1.0f / Ri, invRe = 1.0f / Re;
    const float invCi = 1.0f / Ci, invCe = 1.0f / Ce;

    // x_{t+1} = M x_t + u_t  with  M = [[1-a, a],[b, 1-b-c]]
    const float a = DT * invRi * invCi;
    const float b = DT * invRi * invCe;
    const float c = DT * invRe * invCe;
    const float kH  = DT * invCi * cooling * Hg;      // * hvac_u
    const float kAi = DT * invCi * Ai;                // * solar  (into u0)
    const float kTo = DT * invCe * invRe;             // * t_out  (into u1)
    const float kAe = DT * invCe * Ae;                // * solar  (into u1)
    const float m00 = 1.0f - a, m01 = a;
    const float m10 = b,        m11 = 1.0f - b - c;

    const float* rowp = in + (size_t)row * (T_STEPS * 4);
    __builtin_prefetch(rowp, 0, 1);                   // global_prefetch_b8

    // ---- Phase A: stage the whole row into LDS ----
#if USE_ASYNC_LDS
    // Fire-and-forget global->LDS DMA: 512B per wave-instruction, no VGPR staging.
    for (int k = lane; k < T_STEPS; k += 32) {
        __builtin_amdgcn_global_load_async_to_lds_b128(
            (gvec_ptr)(rowp + 4 * (size_t)k),
            (lvec_ptr)&s_raw[wave][k],
            0, 0);
    }
#if __has_builtin(__builtin_amdgcn_s_wait_asynccnt)
    __builtin_amdgcn_s_wait_asynccnt(0);
#else
    asm volatile("s_wait_asynccnt 0x0" ::: "memory");
#endif
    asm volatile("" ::: "memory");   // keep LDS reads below the wait
#else
    for (int k = lane; k < T_STEPS; k += 32) {
        s_raw[wave][k] = ((const float4*)rowp)[k];    // coalesced b128 load + ds_store
    }
    // wave-local producer->consumer: DS ops are in-order within a wave.
#endif

    // ---- Phase B: per-lane chunk affine offset q_l (start from x = 0) ----
    const int base = lane * CHUNK;
    float s0 = 0.0f, s1 = 0.0f;
    for (int j = 0; j < CHUNK; ++j) {
        float4 v = s_raw[wave][base + j];
        float u0 = kAi * v.w + kH * v.z;
        float u1 = kTo * v.y + kAe * v.w;
        float d0 = a * (s1 - s0) + u0;
        float d1 = b * (s0 - s1) - c * s1 + u1;
        s0 += d0;
        s1 += d1;
    }
    // P = M^CHUNK (uniform)
    float p00 = 1.0f, p01 = 0.0f, p10 = 0.0f, p11 = 1.0f;
    for (int i = 0; i < CHUNK; ++i) {
        float n00 = m00 * p00 + m01 * p10;
        float n01 = m00 * p01 + m01 * p11;
        float n10 = m10 * p00 + m11 * p10;
        float n11 = m10 * p01 + m11 * p11;
        p00 = n00; p01 = n01; p10 = n10; p11 = n11;
    }

    // ---- Phase C: inclusive log-scan of affine maps across 32 lanes ----
    float A00 = p00, A01 = p01, A10 = p10, A11 = p11, V0 = s0, V1 = s1;
    for (int d = 1; d < 32; d <<= 1) {
        float r00 = __shfl_up(A00, d), r01 = __shfl_up(A01, d);
        float r10 = __shfl_up(A10, d), r11 = __shfl_up(A11, d);
        float rv0 = __shfl_up(V0, d),  rv1 = __shfl_up(V1, d);
        if (lane >= d) {
            // new = cur o recv  (recv applied first)
            float n00 = A00 * r00 + A01 * r10;
            float n01 = A00 * r01 + A01 * r11;
            float n10 = A10 * r00 + A11 * r10;
            float n11 = A10 * r01 + A11 * r11;
            float nv0 = A00 * rv0 + A01 * rv1 + V0;
            float nv1 = A10 * rv0 + A11 * rv1 + V1;
            A00 = n00; A01 = n01; A10 = n10; A11 = n11; V0 = nv0; V1 = nv1;
        }
    }
    // exclusive shift: lane l gets composition of chunks [0 .. l-1]
    float e00 = __shfl_up(A00, 1), e01 = __shfl_up(A01, 1);
    float e10 = __shfl_up(A10, 1), e11 = __shfl_up(A11, 1);
    float ev0 = __shfl_up(V0, 1),  ev1 = __shfl_up(V1, 1);
    if (lane == 0) { e00 = 1.0f; e01 = 0.0f; e10 = 0.0f; e11 = 1.0f; ev0 = 0.0f; ev1 = 0.0f; }

    const float tin0  = s_raw[wave][0].x;
    const float tout0 = s_raw[wave][0].y;
    const float tenv0 = 0.5f * (tin0 + tout0);
    float x0 = e00 * tin0 + e01 * tenv0 + ev0;   // state entering step `base`
    float x1 = e10 * tin0 + e11 * tenv0 + ev1;

    // ---- Phase D: replay own chunk, stage t_in outputs in LDS ----
    for (int j = 0; j < CHUNK; ++j) {
        float4 v = s_raw[wave][base + j];
        float u0 = kAi * v.w + kH * v.z;
        float u1 = kTo * v.y + kAe * v.w;
        float d0 = a * (x1 - x0) + u0;
        float d1 = b * (x0 - x1) - c * x1 + u1;
        x0 += d0;
        x1 += d1;
        s_o[wave][base + j] = x0;
    }

    // ---- Phase E: coalesced contiguous global stores ----
    float* orow = out + (size_t)row * T_STEPS;
    for (int k = lane; k < T_STEPS; k += 32) {
        orow[k] = s_o[wave][k];
    }
}

extern "C" void kernel_launch(void* const* d_in, const int* in_sizes, int n_in,
                              void* d_out, int out_size, void* d_ws, size_t ws_size,
                              hipStream_t stream) {
    (void)n_in; (void)out_size; (void)d_ws; (void)ws_size;
    const float* in  = (const float*)d_in[0];
    const float* rRi = (const float*)d_in[1];
    const float* rRe = (const float*)d_in[2];
    const float* rCi = (const float*)d_in[3];
    const float* rCe = (const float*)d_in[4];
    const float* rAi = (const float*)d_in[5];
    const float* rAe = (const float*)d_in[6];
    const float* rHg = (const float*)d_in[7];
    float* out = (float*)d_out;

    const int B = in_sizes[0] / (T_STEPS * 4);
    const int grid = (B + ROWS_PER_BLOCK - 1) / ROWS_PER_BLOCK;

    rc2_affine_scan_kernel<<<grid, ROWS_PER_BLOCK * 32, 0, stream>>>(
        in, rRi, rRe, rCi, rCe, rAi, rAe, rHg, out, B);
}